// SwinMambaBlock_65360812311127
// MI455X (gfx1250) — compile-verified
//
#include <hip/hip_runtime.h>
#include <hip/hip_bf16.h>
#include <stdint.h>

typedef __attribute__((ext_vector_type(2))) float v2f;
typedef __attribute__((ext_vector_type(8))) float v8f;

#define T_TOK   12544           // B*H*W
#define SEQ_L   3136            // H*W
#define NBATCH  4
#define CDIM    128
#define DINNER  256
#define DSTATE  16
#define WSZ     7
#define NWIN    49

enum { ACT_NONE = 0, ACT_SILU, ACT_GELU, ACT_SIGMOID, ACT_SOFTPLUS };

template <int ACT>
__device__ __forceinline__ float act_apply(float v) {
    if (ACT == ACT_SILU)     return v / (1.f + __expf(-v));
    if (ACT == ACT_GELU)     return 0.5f * v * (1.f + erff(v * 0.70710678118654752f));
    if (ACT == ACT_SIGMOID)  return 1.f / (1.f + __expf(-v));
    if (ACT == ACT_SOFTPLUS) return (v > 20.f) ? v : log1pf(__expf(v));
    return v;
}

// Async 16-byte copy global -> LDS (CDNA5 GLOBAL_LOAD_ASYNC_TO_LDS_B128, ASYNCcnt)
__device__ __forceinline__ void async_copy_b128(uint32_t lds_off, const float* gsrc) {
    asm volatile("global_load_async_to_lds_b128 %0, %1, off"
                 :: "v"(lds_off), "v"((uint64_t)(uintptr_t)gsrc)
                 : "memory");
}
__device__ __forceinline__ void wait_async_le3() {
    asm volatile("s_wait_asynccnt 0x3" ::: "memory");
}
__device__ __forceinline__ void wait_async_zero() {
    asm volatile("s_wait_asynccnt 0x0" ::: "memory");
}

// ---------------------------------------------------------------------------
// D = act(A[M,K] @ W[N,K]^T + bias + resid) with V_WMMA_F32_16X16X4_F32.
// 256 threads = 8 waves -> 32(M) x 64(N) block tile; K staged via LDS in
// 32-wide chunks. ASYNC=true: double-buffered GLOBAL_LOAD_ASYNC_TO_LDS_B128
// (requires K%32==0, tiles fully in-bounds). ASYNC=false: guarded sync path
// with zero padding for ragged N/K.
// Fragment layout (ISA 7.12.2):
//   A (16x4): lanes 0-15 row=lid hold K=(k,k+1); lanes 16-31 hold (k+2,k+3)
//   B (4x16): same split, col=lid
//   D: VGPR r -> row r (lanes<16) / r+8 (lanes>=16), col=lid
// LDS row stride 36 floats: 16B-aligned rows for b128, conflict-free for 16
// concurrent row reads (36*r mod 64 distinct for r=0..15).
// ---------------------------------------------------------------------------
#define GEMM_KC 32
#define LDSPAD  36

template <int ACT, bool HAS_BIAS, bool HAS_RES, bool ASYNC>
__global__ void __launch_bounds__(256)
gemm_wmma(const float* __restrict__ A, int lda,
          const float* __restrict__ W, int ldw,
          const float* __restrict__ bias,
          const float* __restrict__ resid, int ldr,
          float* __restrict__ D, int ldd,
          int M, int N, int K, int Wrows)
{
    __shared__ float sA[2][32][LDSPAD];
    __shared__ float sW[2][64][LDSPAD];

    const int tid  = threadIdx.x;
    const int wave = tid >> 5;
    const int lane = tid & 31;
    const int hsel = lane >> 4;
    const int lid  = lane & 15;

    const int nbn = (N + 63) >> 6;
    const int gm0 = (blockIdx.x / nbn) * 32;
    const int gn0 = (blockIdx.x % nbn) * 64;
    const int tm  = wave >> 2;       // 0..1
    const int tn  = wave & 3;        // 0..3

    v8f acc = {};
    const int nk = (K + GEMM_KC - 1) / GEMM_KC;

    auto compute = [&](const float (*cA)[LDSPAD], const float (*cW)[LDSPAD]) {
        const float* ar = cA[tm * 16 + lid];
        const float* wr = cW[tn * 16 + lid];
#pragma unroll
        for (int k = 0; k < GEMM_KC; k += 4) {
            const int ka = k + 2 * hsel;
            v2f a, b;
            a.x = ar[ka]; a.y = ar[ka + 1];
            b.x = wr[ka]; b.y = wr[ka + 1];
            acc = __builtin_amdgcn_wmma_f32_16x16x4_f32(
                false, a, false, b, (short)0, acc, false, false);
        }
    };

    if (ASYNC) {
        // each thread: 1 b128 for A (32x32) + 2 b128 for W (64x32) per stage
        auto issue_stage = [&](int buf, int k0) {
            const int ra = tid >> 3, ca = (tid & 7) << 2;
            async_copy_b128((uint32_t)(uintptr_t)&sA[buf][ra][ca],
                            A + (size_t)(gm0 + ra) * lda + k0 + ca);
#pragma unroll
            for (int i = 0; i < 2; ++i) {
                const int e = tid + (i << 8);
                const int rw = e >> 3, cw = (e & 7) << 2;
                async_copy_b128((uint32_t)(uintptr_t)&sW[buf][rw][cw],
                                W + (size_t)(gn0 + rw) * ldw + k0 + cw);
            }
        };
        issue_stage(0, 0);
        for (int kc = 0; kc < nk; ++kc) {
            if (kc + 1 < nk) {
                issue_stage((kc + 1) & 1, (kc + 1) * GEMM_KC); // overlap next chunk
                wait_async_le3();   // in-order: <=3 outstanding => chunk kc landed
            } else {
                wait_async_zero();
            }
            __syncthreads();        // all waves' copies for chunk kc visible
            compute(sA[kc & 1], sW[kc & 1]);
            __syncthreads();        // done reading before buffer is rewritten
        }
    } else {
        for (int kc = 0; kc < nk; ++kc) {
            const int k0 = kc * GEMM_KC;
            for (int e = tid; e < 32 * GEMM_KC; e += 256) {
                int r = e >> 5, kk = e & 31;
                int gm = gm0 + r, gk = k0 + kk;
                sA[0][r][kk] = (gm < M && gk < K) ? A[(size_t)gm * lda + gk] : 0.f;
            }
            for (int e = tid; e < 64 * GEMM_KC; e += 256) {
                int r = e >> 5, kk = e & 31;
                int gn = gn0 + r, gk = k0 + kk;
                sW[0][r][kk] = (gn < Wrows && gk < K) ? W[(size_t)gn * ldw + gk] : 0.f;
            }
            if (kc + 1 < nk) {
                int r = tid >> 3;
                if (gm0 + r < M) __builtin_prefetch(&A[(size_t)(gm0 + r) * lda + k0 + GEMM_KC], 0, 1);
            }
            __syncthreads();
            compute(sA[0], sW[0]);
            __syncthreads();
        }
    }

    const int n = gn0 + tn * 16 + lid;
    if (n < N) {
        const float bv = HAS_BIAS ? bias[n] : 0.f;
        const int m_base = gm0 + tm * 16 + hsel * 8;
#pragma unroll
        for (int r = 0; r < 8; ++r) {
            int m = m_base + r;
            if (m < M) {
                float v = acc[r] + bv;
                if (HAS_RES) v += resid[(size_t)m * ldr + n];
                D[(size_t)m * ldd + n] = act_apply<ACT>(v);
            }
        }
    }
}

// ---------------------------------------------------------------------------
// LayerNorm over C=128, one wave32 per token (4 tokens / 128-thread block)
// ---------------------------------------------------------------------------
__global__ void __launch_bounds__(128)
layernorm_kernel(const float* __restrict__ x, const float* __restrict__ g,
                 const float* __restrict__ b, float* __restrict__ y)
{
    const int t    = blockIdx.x * 4 + (threadIdx.x >> 5);
    const int lane = threadIdx.x & 31;
    const float* xr = x + (size_t)t * CDIM;

    float v[4]; float s = 0.f;
#pragma unroll
    for (int i = 0; i < 4; ++i) { v[i] = xr[lane + 32 * i]; s += v[i]; }
#pragma unroll
    for (int off = 16; off > 0; off >>= 1) s += __shfl_xor(s, off, 32);
    const float mean = s * (1.f / CDIM);
    float q = 0.f;
#pragma unroll
    for (int i = 0; i < 4; ++i) { float d = v[i] - mean; q += d * d; }
#pragma unroll
    for (int off = 16; off > 0; off >>= 1) q += __shfl_xor(q, off, 32);
    const float rstd = rsqrtf(q * (1.f / CDIM) + 1e-5f);

    float* yr = y + (size_t)t * CDIM;
#pragma unroll
    for (int i = 0; i < 4; ++i) {
        int c = lane + 32 * i;
        yr[c] = (v[i] - mean) * rstd * g[c] + b[c];
    }
}

// ---------------------------------------------------------------------------
// Window attention: one block per (batch, window, head). N=49, hd=32.
// ---------------------------------------------------------------------------
__global__ void __launch_bounds__(64)
attn_kernel(const float* __restrict__ qkv, const float* __restrict__ rpb,
            float* __restrict__ out)
{
    const int blk  = blockIdx.x;
    const int head = blk & 3;
    const int wid  = blk >> 2;
    const int ww   = wid & 7;
    const int wh   = (wid >> 3) & 7;
    const int b    = wid >> 6;

    __shared__ float sq[NWIN][33], sk[NWIN][33], sv[NWIN][33];
    __shared__ float ss[NWIN][NWIN + 1];
    __shared__ int   stok[NWIN];

    const int tid = threadIdx.x;
    if (tid < NWIN) {
        int ih = tid / WSZ, iw = tid % WSZ;
        stok[tid] = b * SEQ_L + (wh * WSZ + ih) * 56 + (ww * WSZ + iw);
    }
    __syncthreads();

    const float scale = 0.17677669529663687f;   // 1/sqrt(32)
    for (int e = tid; e < NWIN * 32; e += 64) {
        int i = e >> 5, d = e & 31;
        size_t base = (size_t)stok[i] * 384 + head * 32 + d;
        sq[i][d] = qkv[base] * scale;
        sk[i][d] = qkv[base + 128];
        sv[i][d] = qkv[base + 256];
    }
    __syncthreads();

    for (int e = tid; e < NWIN * NWIN; e += 64) {
        int i = e / NWIN, j = e % NWIN;
        float s = 0.f;
#pragma unroll
        for (int d = 0; d < 32; ++d) s += sq[i][d] * sk[j][d];
        int yi = i / WSZ, xi = i % WSZ, yj = j / WSZ, xj = j % WSZ;
        int ridx = (yi - yj + 6) * 13 + (xi - xj + 6);
        ss[i][j] = s + rpb[ridx * 4 + head];
    }
    __syncthreads();

    if (tid < NWIN) {
        float mx = -1e30f;
        for (int j = 0; j < NWIN; ++j) mx = fmaxf(mx, ss[tid][j]);
        float sum = 0.f;
        for (int j = 0; j < NWIN; ++j) { float e = __expf(ss[tid][j] - mx); ss[tid][j] = e; sum += e; }
        float inv = 1.f / sum;
        for (int j = 0; j < NWIN; ++j) ss[tid][j] *= inv;
    }
    __syncthreads();

    for (int e = tid; e < NWIN * 32; e += 64) {
        int i = e >> 5, d = e & 31;
        float s = 0.f;
        for (int j = 0; j < NWIN; ++j) s += ss[i][j] * sv[j][d];
        out[(size_t)stok[i] * CDIM + head * 32 + d] = s;
    }
}

// ---------------------------------------------------------------------------
// Depthwise causal conv1d(4) over L per batch + SiLU. xc = xz[:, 0:256].
// ---------------------------------------------------------------------------
__global__ void conv_silu_kernel(const float* __restrict__ xz,
                                 const float* __restrict__ cw,
                                 const float* __restrict__ cb,
                                 float* __restrict__ xconv)
{
    int idx = blockIdx.x * blockDim.x + threadIdx.x;
    if (idx >= T_TOK * DINNER) return;
    int c = idx & 255;
    int t = idx >> 8;
    int l = t % SEQ_L;
    float s = cb[c];
#pragma unroll
    for (int j = 0; j < 4; ++j) {
        int ls = l - 3 + j;
        if (ls >= 0) s += cw[c * 4 + j] * xz[(size_t)(t - l + ls) * 512 + c];
    }
    xconv[idx] = s / (1.f + __expf(-s));
}

// ---------------------------------------------------------------------------
// Selective scan: one block per batch; thread di holds h[16] + A row in VGPRs;
// B(t)/C(t) staged through LDS. Fused epilogue: (y + D*xc)*silu(z).
// dbl layout [T,48]: dt=0..7, B=8..23, C=24..39.
// ---------------------------------------------------------------------------
__global__ void __launch_bounds__(256)
scan_kernel(const float* __restrict__ dtp,
            const float* __restrict__ xconv,
            const float* __restrict__ dbl,
            const float* __restrict__ xz,
            const float* __restrict__ A_log,
            const float* __restrict__ Dp,
            float* __restrict__ yfin)
{
    const int b  = blockIdx.x;
    const int di = threadIdx.x;
    float Aa[DSTATE], h[DSTATE];
#pragma unroll
    for (int s = 0; s < DSTATE; ++s) { Aa[s] = -__expf(A_log[di * DSTATE + s]); h[s] = 0.f; }
    const float dval = Dp[di];

    __shared__ float sB[DSTATE], sC[DSTATE];
    for (int l = 0; l < SEQ_L; ++l) {
        const size_t t = (size_t)b * SEQ_L + l;
        if (di < 32) {
            float v = dbl[t * 48 + 8 + di];
            if (di < 16) sB[di] = v; else sC[di - 16] = v;
        }
        __syncthreads();
        const float dt  = dtp[t * DINNER + di];
        const float xcv = xconv[t * DINNER + di];
        float y = 0.f;
#pragma unroll
        for (int s = 0; s < DSTATE; ++s) {
            float dA = __expf(dt * Aa[s]);
            h[s] = dA * h[s] + (dt * sB[s]) * xcv;
            y += h[s] * sC[s];
        }
        const float zz = xz[t * 512 + 256 + di];
        yfin[t * DINNER + di] = (y + xcv * dval) * (zz / (1.f + __expf(-zz)));
        __syncthreads();
    }
}

__global__ void concat_kernel(const float* __restrict__ a, const float* __restrict__ m,
                              float* __restrict__ comb)
{
    int idx = blockIdx.x * blockDim.x + threadIdx.x;
    if (idx >= T_TOK * 256) return;
    int c = idx & 255, t = idx >> 8;
    comb[idx] = (c < CDIM) ? a[(size_t)t * CDIM + c] : m[(size_t)t * CDIM + c - CDIM];
}

__global__ void fuse_kernel(const float* __restrict__ x, const float* __restrict__ g,
                            const float* __restrict__ a, const float* __restrict__ m,
                            float* __restrict__ x1)
{
    int idx = blockIdx.x * blockDim.x + threadIdx.x;
    if (idx >= T_TOK * CDIM) return;
    float gv = g[idx];
    x1[idx] = x[idx] + gv * a[idx] + (1.f - gv) * m[idx];
}

// ---------------------------------------------------------------------------
extern "C" void kernel_launch(void* const* d_in, const int* in_sizes, int n_in,
                              void* d_out, int out_size, void* d_ws, size_t ws_size,
                              hipStream_t stream)
{
    const float* x         = (const float*)d_in[0];
    const float* ln1_g     = (const float*)d_in[1];
    const float* ln1_b     = (const float*)d_in[2];
    const float* qkv_w     = (const float*)d_in[3];
    const float* qkv_b     = (const float*)d_in[4];
    const float* rpb       = (const float*)d_in[5];
    const float* proj_w    = (const float*)d_in[6];
    const float* proj_b    = (const float*)d_in[7];
    const float* in_proj_w = (const float*)d_in[8];
    const float* in_proj_b = (const float*)d_in[9];
    const float* conv_w    = (const float*)d_in[10];
    const float* conv_b    = (const float*)d_in[11];
    const float* x_proj_w  = (const float*)d_in[12];
    const float* dt_proj_w = (const float*)d_in[13];
    const float* dt_proj_b = (const float*)d_in[14];
    const float* A_log     = (const float*)d_in[15];
    const float* Dp        = (const float*)d_in[16];
    const float* out_proj_w= (const float*)d_in[17];
    const float* out_proj_b= (const float*)d_in[18];
    const float* gate_w    = (const float*)d_in[19];
    const float* gate_b    = (const float*)d_in[20];
    const float* ln2_g     = (const float*)d_in[21];
    const float* ln2_b     = (const float*)d_in[22];
    const float* mlp_w1    = (const float*)d_in[23];
    const float* mlp_b1    = (const float*)d_in[24];
    const float* mlp_w2    = (const float*)d_in[25];
    const float* mlp_b2    = (const float*)d_in[26];

    float* ws = (float*)d_ws;
    size_t off = 0;
    auto alloc = [&](size_t n) { float* p = ws + off; off += n; return p; };

    const size_t T = T_TOK;
    float* xn       = alloc(T * 128);
    float* qkv      = alloc(T * 384);   // reused as `comb` after attention
    float* aw       = alloc(T * 128);
    float* attn_out = alloc(T * 128);
    float* xz       = alloc(T * 512);   // reused as MLP hidden after scan
    float* xconv    = alloc(T * 256);
    float* dbl      = alloc(T * 48);
    float* dtp      = alloc(T * 256);
    float* yfin     = alloc(T * 256);
    float* mamba    = alloc(T * 128);
    float* gbuf     = alloc(T * 128);
    float* x1       = alloc(T * 128);
    float* h2       = alloc(T * 128);
    float* comb = qkv;
    float* hmid = xz;
    float* outp = (float*)d_out;

    auto ggrid = [](int M, int N) { return dim3(((M + 31) / 32) * ((N + 63) / 64)); };

    // 1. LN1
    layernorm_kernel<<<T_TOK / 4, 128, 0, stream>>>(x, ln1_g, ln1_b, xn);
    // 2. QKV
    gemm_wmma<ACT_NONE, true, false, true><<<ggrid(T_TOK, 384), 256, 0, stream>>>(
        xn, 128, qkv_w, 128, qkv_b, nullptr, 0, qkv, 384, T_TOK, 384, 128, 384);
    // 3. Window attention
    attn_kernel<<<1024, 64, 0, stream>>>(qkv, rpb, aw);
    // 4. Attention projection
    gemm_wmma<ACT_NONE, true, false, true><<<ggrid(T_TOK, 128), 256, 0, stream>>>(
        aw, 128, proj_w, 128, proj_b, nullptr, 0, attn_out, 128, T_TOK, 128, 128, 128);
    // 5. Mamba in_proj -> xz = [xc | z]
    gemm_wmma<ACT_NONE, true, false, true><<<ggrid(T_TOK, 512), 256, 0, stream>>>(
        xn, 128, in_proj_w, 128, in_proj_b, nullptr, 0, xz, 512, T_TOK, 512, 128, 512);
    // 6. Causal depthwise conv + SiLU
    conv_silu_kernel<<<(T_TOK * 256 + 255) / 256, 256, 0, stream>>>(xz, conv_w, conv_b, xconv);
    // 7. x_proj (ragged N=40 -> sync path, ld=48)
    gemm_wmma<ACT_NONE, false, false, false><<<ggrid(T_TOK, 40), 256, 0, stream>>>(
        xconv, 256, x_proj_w, 256, nullptr, nullptr, 0, dbl, 48, T_TOK, 40, 256, 40);
    // 8. dt_proj + softplus (ragged K=8 -> sync path)
    gemm_wmma<ACT_SOFTPLUS, true, false, false><<<ggrid(T_TOK, 256), 256, 0, stream>>>(
        dbl, 48, dt_proj_w, 8, dt_proj_b, nullptr, 0, dtp, 256, T_TOK, 256, 8, 256);
    // 9. Selective scan + fused epilogue
    scan_kernel<<<NBATCH, 256, 0, stream>>>(dtp, xconv, dbl, xz, A_log, Dp, yfin);
    // 10. out_proj
    gemm_wmma<ACT_NONE, true, false, true><<<ggrid(T_TOK, 128), 256, 0, stream>>>(
        yfin, 256, out_proj_w, 256, out_proj_b, nullptr, 0, mamba, 128, T_TOK, 128, 256, 128);
    // 11. concat + learned gate + fuse
    concat_kernel<<<(T_TOK * 256 + 255) / 256, 256, 0, stream>>>(attn_out, mamba, comb);
    gemm_wmma<ACT_SIGMOID, true, false, true><<<ggrid(T_TOK, 128), 256, 0, stream>>>(
        comb, 256, gate_w, 256, gate_b, nullptr, 0, gbuf, 128, T_TOK, 128, 256, 128);
    fuse_kernel<<<(T_TOK * 128 + 255) / 256, 256, 0, stream>>>(x, gbuf, attn_out, mamba, x1);
    // 12. LN2 + MLP + residual into d_out
    layernorm_kernel<<<T_TOK / 4, 128, 0, stream>>>(x1, ln2_g, ln2_b, h2);
    gemm_wmma<ACT_GELU, true, false, true><<<ggrid(T_TOK, 512), 256, 0, stream>>>(
        h2, 128, mlp_w1, 128, mlp_b1, nullptr, 0, hmid, 512, T_TOK, 512, 128, 512);
    gemm_wmma<ACT_NONE, true, true, true><<<ggrid(T_TOK, 128), 256, 0, stream>>>(
        hmid, 512, mlp_w2, 512, mlp_b2, x1, 128, outp, 128, T_TOK, 128, 512, 128);
}